// CustomCrossEntropyLoss_5368709120380
// MI455X (gfx1250) — compile-verified
//
#include <hip/hip_runtime.h>
#include <hip/hip_bf16.h>
#include <stdint.h>

// Problem constants from the reference: N = 1024 classes (dim = 10 bits).
#define N_CLASSES 1024
#define TPB       256
#define EPT       (N_CLASSES / TPB)   // 4 elements per thread
#define NWAVES    (TPB / 32)          // 8 wave32s per block

typedef __attribute__((ext_vector_type(4))) float v4f;  // raw clang vector: OK for nontemporal builtin

// 6^p for p in [0,10]; all exactly representable in fp32 (6^10 < 2^26).
__device__ const float g_p6[11] = {
    1.0f, 6.0f, 36.0f, 216.0f, 1296.0f, 7776.0f, 46656.0f,
    279936.0f, 1679616.0f, 10077696.0f, 60466176.0f
};

__device__ inline void wave_argmax(float& v, int& i) {
    #pragma unroll
    for (int m = 16; m >= 1; m >>= 1) {
        float ov = __shfl_xor(v, m, 32);
        int   oi = __shfl_xor(i, m, 32);
        if (ov > v || (ov == v && oi < i)) { v = ov; i = oi; }
    }
}

__device__ inline float wave_sum(float v) {
    #pragma unroll
    for (int m = 16; m >= 1; m >>= 1) v += __shfl_xor(v, m, 32);
    return v;
}

// One block per row. Phase 1: argmax(target row). Phase 2: bit-Hamming
// weighted sum over the y row, staged into LDS via CDNA5 async global->LDS
// DMA issued up front so the fetch overlaps the argmax phase entirely.
__global__ __launch_bounds__(TPB) void ce_bitham_row_kernel(
    const float* __restrict__ y_true,
    const float* __restrict__ target,
    float* __restrict__ partial,
    float inv_B, float inv_BN)
{
    __shared__ float ybuf[N_CLASSES];   // 4 KB: y_true row staged by async DMA
    __shared__ float rval[NWAVES];
    __shared__ int   ridx[NWAVES];
    __shared__ float p6[16];            // 6^p LUT

    const int tid  = threadIdx.x;
    const int wave = tid >> 5;
    const int lane = tid & 31;
    const long long base = (long long)blockIdx.x * N_CLASSES;

    // ---- kick off async copy of this row of y_true into LDS (16B/lane) ----
    // CDNA5: GLOBAL_LOAD_ASYNC_TO_LDS_B128, tracked with ASYNCcnt.
    {
        uint32_t lds_off = (uint32_t)(uintptr_t)(&ybuf[tid * EPT]);
        const float* gp = y_true + base + tid * EPT;
        asm volatile("global_load_async_to_lds_b128 %0, %1, off"
                     :: "v"(lds_off), "v"(gp)
                     : "memory");
    }

    if (tid < 11) p6[tid] = g_p6[tid];

    // ---- phase 1: argmax over target row, first occurrence wins ----
    // Stream target non-temporally: single-use data, don't pollute L2.
    v4f tv = __builtin_nontemporal_load((const v4f*)(target + base) + tid);
    float bv = tv.x; int bi = tid * EPT;
    if (tv.y > bv) { bv = tv.y; bi = tid * EPT + 1; }
    if (tv.z > bv) { bv = tv.z; bi = tid * EPT + 2; }
    if (tv.w > bv) { bv = tv.w; bi = tid * EPT + 3; }

    wave_argmax(bv, bi);                       // in-register, wave32
    if (lane == 0) { rval[wave] = bv; ridx[wave] = bi; }
    __syncthreads();

    // Every thread folds the 8 wave partials (broadcast LDS reads, no 2nd barrier).
    float fv = rval[0]; int t = ridx[0];
    #pragma unroll
    for (int k = 1; k < NWAVES; ++k) {
        float ov = rval[k]; int oi = ridx[k];
        if (ov > fv || (ov == fv && oi < t)) { fv = ov; t = oi; }
    }

    // ---- drain our wave's async DMA, then sync whole workgroup ----
    asm volatile("s_wait_asynccnt 0x0" ::: "memory");
    __syncthreads();

    // ---- phase 2: sum_c w_table[t ^ c] * y[b,c] from LDS ----
    float acc = 0.0f;
    #pragma unroll
    for (int i = 0; i < EPT; ++i) {
        int c = tid * EPT + i;
        int x = c ^ t;
        float w = x ? p6[__popc(x)] : 0.0f;
        acc += w * ybuf[c];
    }
    acc = wave_sum(acc);
    if (lane == 0) rval[wave] = acc;
    __syncthreads();

    if (tid == 0) {
        float pt = 0.0f;
        #pragma unroll
        for (int k = 0; k < NWAVES; ++k) pt += rval[k];
        float lg = __logf(ybuf[t] + 1e-8f);    // CE term: y[b,t] from LDS
        partial[blockIdx.x] = pt * inv_BN - lg * inv_B;
    }
}

// Deterministic final reduction of per-row partials -> scalar loss.
__global__ __launch_bounds__(1024) void final_reduce_kernel(
    const float* __restrict__ partial, float* __restrict__ out, int n)
{
    __shared__ double sm[32];
    double s = 0.0;
    for (int i = threadIdx.x; i < n; i += 1024) s += (double)partial[i];
    #pragma unroll
    for (int m = 16; m >= 1; m >>= 1) s += __shfl_xor(s, m, 32);
    const int wave = threadIdx.x >> 5, lane = threadIdx.x & 31;
    if (lane == 0) sm[wave] = s;
    __syncthreads();
    if (threadIdx.x == 0) {
        double tot = 0.0;
        #pragma unroll
        for (int k = 0; k < 32; ++k) tot += sm[k];
        out[0] = (float)tot;
    }
}

extern "C" void kernel_launch(void* const* d_in, const int* in_sizes, int n_in,
                              void* d_out, int out_size, void* d_ws, size_t ws_size,
                              hipStream_t stream) {
    const float* y_true = (const float*)d_in[0];
    const float* target = (const float*)d_in[1];
    float* out = (float*)d_out;
    float* partial = (float*)d_ws;

    const int B = in_sizes[0] / N_CLASSES;   // 65536 for the reference shapes
    const float inv_B  = 1.0f / (float)B;                          // 2^-16, exact
    const float inv_BN = 1.0f / ((float)B * (float)N_CLASSES);     // 2^-26, exact

    ce_bitham_row_kernel<<<B, TPB, 0, stream>>>(y_true, target, partial, inv_B, inv_BN);
    final_reduce_kernel<<<1, 1024, 0, stream>>>(partial, out, B);
}